// VoxelWithPointProjection_22256520528522
// MI455X (gfx1250) — compile-verified
//
#include <hip/hip_runtime.h>

// ---------------------------------------------------------------------------
// VoxelWithPointProjection fused gather kernel for MI455X (gfx1250).
//
// out[n,c] = voxel_feats[n,c] + sum_k mask[k,n] * img[k, batch[n], c, y[k,n], x[k,n]]
//
// Memory-bound (~390 MB HBM traffic -> ~17us at 23.3 TB/s). No matmul
// structure (per-voxel gathered operand), so WMMA does not apply; instead we
// exercise the CDNA5 Tensor Data Mover to overlap the streaming voxel_feats
// read (LDS staging via one TDM descriptor per block) with the random
// img_feats gathers, synchronized with s_wait_tensorcnt + workgroup barrier.
// ---------------------------------------------------------------------------

namespace {
constexpr int kCams = 6;
constexpr int kB    = 2;
constexpr int kC    = 64;
constexpr int kH    = 232;
constexpr int kW    = 400;
constexpr int kVoxPerBlock  = 8;                    // 8 voxels * 64 ch = 512 thr
constexpr int kBlockThreads = kVoxPerBlock * kC;    // 16 wave32s
}

typedef __attribute__((ext_vector_type(4))) unsigned int u32x4;
typedef __attribute__((ext_vector_type(8))) int          i32x8;
typedef __attribute__((ext_vector_type(4))) int          i32x4;

__global__ __launch_bounds__(kBlockThreads)
void voxel_fuse_kernel(const float* __restrict__ img,    // (6,2,64,232,400)
                       const float* __restrict__ vox,    // (N,64)
                       const int*   __restrict__ bidx,   // (N,)
                       const int*   __restrict__ gy,     // (6,N)
                       const int*   __restrict__ gx,     // (6,N)
                       const int*   __restrict__ pm,     // (6,N)
                       float*       __restrict__ out,    // (N,64)
                       int N)
{
  __shared__ float tile[kBlockThreads];

  const int n0 = blockIdx.x * kVoxPerBlock;

  // ---- Wave 0: kick off TDM DMA of this block's voxel_feats tile into LDS.
  // TENSORcnt is per-wave, so the issuing wave alone waits on it (below),
  // then __syncthreads() publishes the LDS tile to all 16 waves.
  if ((threadIdx.x >> 5) == 0) {
    const unsigned long long gaddr =
        (unsigned long long)(const void*)(vox + (unsigned long long)n0 * kC);
    const unsigned int lds_addr = (unsigned int)(unsigned long long)(void*)&tile[0];
    unsigned int elems = (unsigned int)kBlockThreads;           // 512 floats (2 KB)
    const unsigned int remain = (unsigned int)(N - n0) * kC;
    if (remain < elems) elems = remain;                          // tail guard

    // D# group 0: count=1 | lds_addr | global_addr[56:0] | type=2 ("image")
    u32x4 g0;
    g0[0] = 1u;                                                  // count=1, user desc
    g0[1] = lds_addr;                                            // LDS byte address
    g0[2] = (unsigned int)(gaddr & 0xFFFFFFFFull);               // global_addr[31:0]
    g0[3] = (unsigned int)((gaddr >> 32) & 0x1FFFFFFull)         // global_addr[56:32]
          | (2u << 30);                                          // type=2

    // D# group 1: data_size=4B; 2D tensor elems x 1; tile = elems x 1
    i32x8 g1;
    g1[0] = (int)(2u << 16);                                     // data_size=2 (4 bytes)
    g1[1] = (int)((elems & 0xFFFFu) << 16);                      // tensor_dim0[15:0]
    g1[2] = (int)((elems >> 16) | (1u << 16));                   // tensor_dim0[31:16], tensor_dim1=1
    g1[3] = (int)((elems & 0xFFFFu) << 16);                      // tile_dim0 = elems
    g1[4] = 1;                                                   // tile_dim1 = 1
    g1[5] = (int)elems;                                          // tensor_dim0_stride[31:0]
    g1[6] = (int)((elems & 0xFFFFu) << 16);                      // tensor_dim1_stride[15:0]
    g1[7] = 0;

    const i32x4 z4 = {0, 0, 0, 0};                               // groups 2/3: <=2D tensor
    const i32x8 z8 = {0, 0, 0, 0, 0, 0, 0, 0};                   // unused extra group
    __builtin_amdgcn_tensor_load_to_lds(g0, g1, z4, z4, z8, /*cpol=*/0);
  }

  // ---- All waves: random masked gathers (overlapped with the TDM copy).
  const int local = (int)threadIdx.x;
  const int c     = local & (kC - 1);
  const int n     = n0 + (local >> 6);     // wave-uniform: 32 lanes share a voxel
  float acc = 0.0f;

  if (n < N) {
    const int b = bidx[n];                 // wave-uniform load
    #pragma unroll
    for (int k = 0; k < kCams; ++k) {
      const long long e = (long long)k * N + n;
      if (pm[e] > 0) {                     // wave-uniform branch: skip dead cams
        const int y = gy[e];
        const int x = gx[e];
        const long long plane = (long long)(k * kB + b) * kC + c;
        acc += img[plane * (long long)(kH * kW) + (long long)y * kW + x];
      }
    }
  }

  // ---- Drain the TDM transfer (wave 0 only), then publish LDS tile.
  if ((threadIdx.x >> 5) == 0) {
    __builtin_amdgcn_s_wait_tensorcnt(0);
  }
  __syncthreads();

  if (n < N) {
    const float r = acc + tile[local];
    __builtin_nontemporal_store(r, &out[(long long)n * kC + c]); // streaming store
  }
}

extern "C" void kernel_launch(void* const* d_in, const int* in_sizes, int n_in,
                              void* d_out, int out_size, void* d_ws, size_t ws_size,
                              hipStream_t stream) {
  const float* img  = (const float*)d_in[0];   // img_feats (6,2,64,232,400) f32
  const float* vox  = (const float*)d_in[1];   // voxel_feats (N,64) f32
  const int*   bidx = (const int*)d_in[2];     // voxel_batch_idx (N,)
  const int*   gy   = (const int*)d_in[3];     // grid_y (6,N)
  const int*   gx   = (const int*)d_in[4];     // grid_x (6,N)
  const int*   pm   = (const int*)d_in[5];     // point_mask (6,N)
  float*       out  = (float*)d_out;           // (N,64) f32

  const int N = in_sizes[2];                   // 200000
  const int grid = (N + kVoxPerBlock - 1) / kVoxPerBlock;

  voxel_fuse_kernel<<<grid, kBlockThreads, 0, stream>>>(img, vox, bidx, gy, gx,
                                                        pm, out, N);
}